// SimClr_loss_66692252172929
// MI455X (gfx1250) — compile-verified
//
#include <hip/hip_runtime.h>
#include <math.h>

#define N_TOT   8192
#define DIMF    128
#define HALF_N  4096
#define INV_T   (1.0f / 0.07f)
#define NEG_BIG (-1.0e30f)

#define PAD       4
#define LDSSTRIDE (DIMF + PAD)          // 132 floats -> conflict-free ds_load_b64
#define ROW_TILES (N_TOT / 16)          // 512 tiles of 16 columns
#define QUARTERS  4
#define TILES_PER_BLOCK (ROW_TILES / QUARTERS)  // 128

typedef __attribute__((ext_vector_type(2))) float v2f;
typedef __attribute__((ext_vector_type(8))) float v8f;

// Pointer types matching __builtin_amdgcn_global_load_async_to_lds_b128:
// param0: v4i in AS1 (global), param1: v4i in AS3 (LDS)
typedef int v4i_t __attribute__((vector_size(16)));
typedef __attribute__((address_space(1))) v4i_t* g_v4i_p;
typedef __attribute__((address_space(3))) v4i_t* l_v4i_p;

// ---- CDNA5 async global -> LDS copy (ASYNCcnt-tracked) ------------------
__device__ __forceinline__ void async_copy_b128(const float* g, float* l) {
#if __has_builtin(__builtin_amdgcn_global_load_async_to_lds_b128)
  __builtin_amdgcn_global_load_async_to_lds_b128((g_v4i_p)g, (l_v4i_p)l, 0, 0);
#else
  l_v4i_p l3 = (l_v4i_p)l;
  asm volatile("global_load_async_to_lds_b128 %0, %1, off"
               :: "v"(l3), "v"(g) : "memory");
#endif
}

__device__ __forceinline__ void wait_asynccnt_le4() {
#if __has_builtin(__builtin_amdgcn_s_wait_asynccnt)
  __builtin_amdgcn_s_wait_asynccnt(4);
#else
  asm volatile("s_wait_asynccnt 0x4" ::: "memory");
#endif
}
__device__ __forceinline__ void wait_asynccnt_0() {
#if __has_builtin(__builtin_amdgcn_s_wait_asynccnt)
  __builtin_amdgcn_s_wait_asynccnt(0);
#else
  asm volatile("s_wait_asynccnt 0x0" ::: "memory");
#endif
}

// Stage one 16-col x 128-K fp32 tile (8KB) into padded LDS, 4 b128 per thread.
__device__ __forceinline__ void tile_async_load(const float* __restrict__ F,
                                                int colBase, float* shbuf,
                                                int tid) {
#pragma unroll
  for (int j = 0; j < 4; ++j) {
    const int q  = tid * 4 + j;     // 512 quads = 16 rows x 32 quads
    const int c  = q >> 5;          // tile row (column of sim)
    const int kq = q & 31;          // float4 within the 128-float row
    async_copy_b128(F + (colBase + c) * DIMF + kq * 4,
                    shbuf + c * LDSSTRIDE + kq * 4);
  }
}

// Online logsumexp update for one 16x16 WMMA f32 C/D tile.
// C/D layout: VGPR r, lane l -> row (r + 8*(l>>4)), col (l&15).
__device__ __forceinline__ void lse_update(const v8f& acc, int isDiagTile,
                                           int lr, int lh,
                                           float m[8], float s[8]) {
#pragma unroll
  for (int r = 0; r < 8; ++r) {
    float v = acc[r] * INV_T;
    if (isDiagTile && (lr == r + 8 * lh)) v = NEG_BIG;   // mask diagonal
    float mn = fmaxf(m[r], v);
    s[r] = s[r] * __expf(m[r] - mn) + __expf(v - mn);
    m[r] = mn;
  }
}

// Block: 64 rows (4 waves x 16 rows) x one column quarter (128 tiles).
// B tiles shared across all 4 waves via double-buffered async LDS staging.
__global__ __launch_bounds__(128) void ntxent_main(const float* __restrict__ F,
                                                   float2* __restrict__ ms) {
  const int quarter = blockIdx.x;       // column quarter 0..3
  const int rowBlk  = blockIdx.y;       // 64-row block 0..127
  const int tid  = threadIdx.x;
  const int wg   = tid >> 5;            // wave = row group
  const int lane = tid & 31;
  const int lr   = lane & 15;
  const int lh   = lane >> 4;
  const int rowBase = rowBlk * 64 + wg * 16;
  const int rowTile = rowBlk * 4 + wg;  // this wave's diagonal tile index

  __shared__ float sh[2][16 * LDSSTRIDE];   // 2 x 8.25KB staging buffers

  // Preload A fragments (16 rows x K=128): lane l, vgpr v holds
  // A[(l&15)][4*kk + 2*(l>>4) + v]
  const float* arow = F + (rowBase + lr) * DIMF + 2 * lh;
  v2f afrag[32];
#pragma unroll
  for (int kk = 0; kk < 32; ++kk)
    afrag[kk] = *(const v2f*)(arow + 4 * kk);

  float m[8], s[8];
#pragma unroll
  for (int r = 0; r < 8; ++r) { m[r] = NEG_BIG; s[r] = 0.0f; }

  const int jt0 = quarter * TILES_PER_BLOCK;
  tile_async_load(F, (jt0 + 0) * 16, sh[0], tid);
  tile_async_load(F, (jt0 + 1) * 16, sh[1], tid);

  for (int i = 0; i < TILES_PER_BLOCK; ++i) {
    const int p = i & 1;
    // oldest in-flight tile (4 async ops/wave) must have landed
    if (i + 1 < TILES_PER_BLOCK) wait_asynccnt_le4();
    else                         wait_asynccnt_0();
    __syncthreads();                      // all waves' portions visible

    const v2f* bp = (const v2f*)(sh[p] + lr * LDSSTRIDE + 2 * lh);
    v2f bfrag[32];
#pragma unroll
    for (int kk = 0; kk < 32; ++kk)
      bfrag[kk] = bp[2 * kk];             // ds_load_b64, conflict-free
    v8f acc = {};
#pragma unroll
    for (int kk = 0; kk < 32; ++kk)
      acc = __builtin_amdgcn_wmma_f32_16x16x4_f32(
          false, afrag[kk], false, bfrag[kk], (short)0, acc, false, false);

    lse_update(acc, (jt0 + i) == rowTile, lr, lh, m, s);

    __syncthreads();                      // buffer p fully consumed
    if (i + 2 < TILES_PER_BLOCK)
      tile_async_load(F, (jt0 + i + 2) * 16, sh[p], tid);
  }

  // merge (m,s) across the 16 lanes of each half-wave (per row-slot r)
#pragma unroll
  for (int r = 0; r < 8; ++r) {
#pragma unroll
    for (int off = 8; off >= 1; off >>= 1) {
      float mo = __shfl_xor(m[r], off, 32);
      float so = __shfl_xor(s[r], off, 32);
      float mn = fmaxf(m[r], mo);
      s[r] = s[r] * __expf(m[r] - mn) + so * __expf(mo - mn);
      m[r] = mn;
    }
  }
  if (lr == 0) {
#pragma unroll
    for (int r = 0; r < 8; ++r) {
      const int row = rowBase + r + 8 * lh;
      ms[row * QUARTERS + quarter] = make_float2(m[r], s[r]);
    }
  }
}

// One thread per row: merge quarters, lse = m + log s, subtract positive.
__global__ __launch_bounds__(256) void ntxent_finalize(const float* __restrict__ F,
                                                       const float2* __restrict__ ms,
                                                       float* __restrict__ partial) {
  const int tid = threadIdx.x;
  const int row = blockIdx.x * 256 + tid;

  float2 v0 = ms[row * QUARTERS];
  float mf = v0.x, sf = v0.y;
#pragma unroll
  for (int q = 1; q < QUARTERS; ++q) {
    float2 vq = ms[row * QUARTERS + q];
    float mn = fmaxf(mf, vq.x);
    sf = sf * __expf(mf - mn) + vq.y * __expf(vq.x - mn);
    mf = mn;
  }
  float lse = mf + __logf(sf);

  const int j = row ^ HALF_N;             // (row + B) % N for B = N/2
  const float4* fi = (const float4*)(F + row * DIMF);
  const float4* fj = (const float4*)(F + j * DIMF);
  float pos = 0.0f;
#pragma unroll
  for (int k = 0; k < DIMF / 4; ++k) {
    float4 a = fi[k], b = fj[k];
    pos += a.x * b.x + a.y * b.y + a.z * b.z + a.w * b.w;
  }
  float term = lse - pos * INV_T;

  __shared__ float lds[8];
#pragma unroll
  for (int off = 16; off >= 1; off >>= 1)
    term += __shfl_xor(term, off, 32);
  if ((tid & 31) == 0) lds[tid >> 5] = term;
  __syncthreads();
  if (tid < 8) {
    float v = lds[tid];
#pragma unroll
    for (int off = 4; off >= 1; off >>= 1)
      v += __shfl_xor(v, off, 32);
    if (tid == 0) partial[blockIdx.x] = v;
  }
}

// Deterministic final sum of 32 block partials -> mean.
__global__ __launch_bounds__(32) void ntxent_sum(const float* __restrict__ partial,
                                                 float* __restrict__ out) {
  float v = partial[threadIdx.x];
#pragma unroll
  for (int off = 16; off >= 1; off >>= 1)
    v += __shfl_xor(v, off, 32);
  if (threadIdx.x == 0) out[0] = v / (float)N_TOT;
}

extern "C" void kernel_launch(void* const* d_in, const int* in_sizes, int n_in,
                              void* d_out, int out_size, void* d_ws, size_t ws_size,
                              hipStream_t stream) {
  const float* F = (const float*)d_in[0];
  float2* ms     = (float2*)d_ws;                         // 8192 x 4 (m,s) pairs
  float* partial = (float*)d_ws + (size_t)N_TOT * QUARTERS * 2;  // 32 floats after

  dim3 grid1(QUARTERS, N_TOT / 64);                        // 4 x 128 blocks
  ntxent_main<<<grid1, 128, 0, stream>>>(F, ms);
  ntxent_finalize<<<N_TOT / 256, 256, 0, stream>>>(F, ms, partial);
  ntxent_sum<<<1, 32, 0, stream>>>(partial, (float*)d_out);
}